// ProdLayer_6554120093765
// MI455X (gfx1250) — compile-verified
//
#include <hip/hip_runtime.h>
#include <stdint.h>

// ProdLayer forward: element_mars[nids_g] = node_mars[cids_g].sum(axis=1)
// Memory-bound gather-sum (~0.8 GB traffic -> ~35 us floor at 23.3 TB/s).
// CDNA5 path: async global->LDS staging (ASYNCcnt), ds_load_b128 reduction,
// non-temporal b128 stores so the 128 MiB output stream doesn't evict the
// gather working set from the 192 MB L2.

typedef float v4f __attribute__((ext_vector_type(4)));  // native vector: OK for nontemporal builtin

template <int C, int R>  // C = arity, R = output rows per tile; C*R == 32 child rows (32 KiB LDS)
__global__ __launch_bounds__(256)
void prod_async_kernel(const float* __restrict__ node_mars,
                       const int* __restrict__ nids,
                       const int* __restrict__ cids,
                       float* __restrict__ out)
{
    static_assert(C * R == 32, "tile must stage exactly 32 child rows");
    __shared__ v4f s_tile[32 * 64];   // 32 child rows x 256 f32 = 32 KiB
    __shared__ int s_cid[32];

    const int t       = threadIdx.x;          // 0..255
    const int rowBase = blockIdx.x * R;

    // Stage the 32 child indices for this tile (row-major: row r, child c -> r*C + c).
    if (t < 32) s_cid[t] = cids[(size_t)rowBase * C + t];
    __syncthreads();

    // Low 32 bits of a generic shared pointer == LDS byte offset within the
    // workgroup allocation (HW adds LDS_BASE for async-to-LDS destinations).
    const uint32_t ldsBase = (uint32_t)(uintptr_t)(&s_tile[0]);

    // Copy phase: 32 child rows * 64 chunks of 16B = 2048 chunks, 8 rounds of 256 lanes.
    // global_load_async_to_lds_b128 vdst(LDS addr), vaddr(32b offset), saddr(base)  [GVS mode]
#pragma unroll
    for (int k = 0; k < 8; ++k) {
        const int chunk       = k * 256 + t;    // 0..2047
        const int slot        = chunk >> 6;     // child-row slot 0..31
        const int within      = chunk & 63;     // 16B chunk within the 1 KiB row
        const uint32_t voff   = ((uint32_t)s_cid[slot] << 10) | ((uint32_t)within << 4);
        const uint32_t laddr  = ldsBase + ((uint32_t)chunk << 4);
        asm volatile("global_load_async_to_lds_b128 %0, %1, %2"
                     :
                     : "v"(laddr), "v"(voff), "s"(node_mars)
                     : "memory");
    }
    // Drain this wave's async copies, then make them visible workgroup-wide.
    asm volatile("s_wait_asynccnt 0x0" ::: "memory");
    __syncthreads();

    // Compute phase: R rows * 64 float4 columns; 256 lanes per round.
#pragma unroll
    for (int k = 0; k < R / 4; ++k) {
        const int item = k * 256 + t;   // 0 .. R*64-1
        const int lr   = item >> 6;     // local output row
        const int col  = item & 63;     // float4 column

        v4f acc = s_tile[(lr * C) * 64 + col];
#pragma unroll
        for (int c = 1; c < C; ++c) {
            acc += s_tile[(lr * C + c) * 64 + col];
        }

        const int drow = nids[rowBase + lr];
        v4f* dst = (v4f*)out + (size_t)drow * 64 + col;
        __builtin_nontemporal_store(acc, dst);   // stream output past L2
    }
}

extern "C" void kernel_launch(void* const* d_in, const int* in_sizes, int n_in,
                              void* d_out, int out_size, void* d_ws, size_t ws_size,
                              hipStream_t stream)
{
    (void)n_in; (void)d_ws; (void)ws_size; (void)out_size;

    const float* node_mars = (const float*)d_in[0];
    // d_in[1] (element_mars, all zeros) is never read: nids0 ∪ nids1 cover every
    // output row exactly once, so the kernels fully overwrite d_out.
    const int* nids0 = (const int*)d_in[2];
    const int* cids0 = (const int*)d_in[3];
    const int* nids1 = (const int*)d_in[4];
    const int* cids1 = (const int*)d_in[5];
    float* out = (float*)d_out;

    const int N0 = in_sizes[2];   // 65536, arity 2
    const int N1 = in_sizes[4];   // 65536, arity 8

    // Arity 2: 16 output rows per tile -> 4096 blocks
    prod_async_kernel<2, 16><<<N0 / 16, 256, 0, stream>>>(node_mars, nids0, cids0, out);
    // Arity 8: 4 output rows per tile -> 16384 blocks
    prod_async_kernel<8, 4><<<N1 / 4, 256, 0, stream>>>(node_mars, nids1, cids1, out);
}